// AdditiveOscillator_30081950941725
// MI455X (gfx1250) — compile-verified
//
#include <hip/hip_runtime.h>
#include <math.h>

typedef float v2f __attribute__((ext_vector_type(2)));
typedef float v8f __attribute__((ext_vector_type(8)));

#define SRATE  24000.0f
#define NYQ    12000.0f
#define HOP    256
#define B_     8
#define C_     256
#define H_     64
#define T_     250
#define TOUT   (T_ * HOP)   // 64000

// align_corners=True mapping: pos = j * (T_in-1)/(T_out-1)
__device__ __forceinline__ float up_pos(int j) {
    return (float)j * ((float)(T_ - 1) / (float)(TOUT - 1));
}

// ---------------------------------------------------------------------------
// Kernel 1: amp[b,h,t] = mask * softplus(W @ z + bias), via f32 WMMA.
// One wave per 16x16 output tile; K-loop of 64 x V_WMMA_F32_16X16X4_F32.
// grid = (ntile=16, mtile=4, batch=8), block = 32 (one wave, EXEC all-1s).
// ---------------------------------------------------------------------------
__global__ void amp_gemm_kernel(const float* __restrict__ z,     // [B,C,T]
                                const float* __restrict__ f0,    // [B,1,T]
                                const float* __restrict__ W,     // [H,C]
                                const float* __restrict__ bias,  // [H]
                                float* __restrict__ amp)         // [B,H,T]
{
    const int lane  = threadIdx.x;     // 0..31
    const int ln    = lane & 15;
    const int hi    = lane >> 4;
    const int ntile = blockIdx.x;      // 0..15 -> t tile
    const int mtile = blockIdx.y;      // 0..3  -> h tile
    const int b     = blockIdx.z;      // 0..7

    const int m  = mtile * 16 + ln;                 // A-matrix row for this lane
    const int t  = ntile * 16 + ln;                 // output column
    const int tc = (t < T_) ? t : (T_ - 1);         // clamp (garbage cols masked at store)

    const float* Wrow = W + (size_t)m * C_;                         // A row
    const float* zcol = z + ((size_t)b * C_) * T_ + tc;             // B column

    v8f acc = {};
    for (int k0 = 0; k0 < C_; k0 += 4) {
        // A 16x4 f32 layout: vgpr0 -> K = k0 + 2*hi, vgpr1 -> K = k0 + 2*hi + 1
        v2f a;
        a.x = Wrow[k0 + 2 * hi];
        a.y = Wrow[k0 + 2 * hi + 1];
        // B 4x16 f32 layout mirrors A: rows striped across lanes within a VGPR
        v2f bv;
        bv.x = zcol[(size_t)(k0 + 2 * hi)     * T_];
        bv.y = zcol[(size_t)(k0 + 2 * hi + 1) * T_];
        acc = __builtin_amdgcn_wmma_f32_16x16x4_f32(false, a, false, bv,
                                                    (short)0, acc, false, false);
    }

    if (t < T_) {
        const float f = f0[b * T_ + t];
        #pragma unroll
        for (int r = 0; r < 8; ++r) {
            const int mrow = mtile * 16 + r + 8 * hi;   // harmonic index 0..63
            float v = acc[r] + bias[mrow];
            // softplus (numerically stable)
            v = (v > 20.0f) ? v : log1pf(__expf(v));
            // anti-alias mask: (n * f0 < SR/2)
            const float nh = (float)(mrow + 1);
            v = (nh * f < NYQ) ? v : 0.0f;
            amp[((size_t)b * H_ + mrow) * T_ + t] = v;
        }
    }
}

// ---------------------------------------------------------------------------
// Kernel 2: per-chunk sums of f0_up/SR + exclusive scan -> chunk phase offsets.
// grid = (8 batches), block = 256 (thread tid owns chunk tid, 250 chunks).
// ---------------------------------------------------------------------------
__global__ void chunk_scan_kernel(const float* __restrict__ f0,
                                  float* __restrict__ chunk_off)  // [B, 250]
{
    const int b   = blockIdx.x;
    const int tid = threadIdx.x;
    __shared__ float s[256];

    float sum = 0.0f;
    if (tid < T_) {
        const float* f = f0 + b * T_;
        for (int i = 0; i < HOP; ++i) {
            const int j = tid * HOP + i;
            float pos = up_pos(j);
            int i0 = (int)pos;  if (i0 > T_ - 1) i0 = T_ - 1;
            int i1 = (i0 + 1 < T_) ? i0 + 1 : T_ - 1;
            float w  = pos - (float)i0;
            float fu = f[i0] * (1.0f - w) + f[i1] * w;
            sum += fu * (1.0f / SRATE);
        }
    }
    s[tid] = sum;
    __syncthreads();
    for (int off = 1; off < 256; off <<= 1) {
        float t = (tid >= off) ? s[tid - off] : 0.0f;
        __syncthreads();
        s[tid] += t;
        __syncthreads();
    }
    if (tid < T_) chunk_off[b * T_ + tid] = s[tid] - sum;   // exclusive prefix
}

// ---------------------------------------------------------------------------
// Kernel 3: intra-chunk phase scan + upsample + sinusoid synthesis.
// grid = (250 chunks, 8 batches), block = 256 (one output sample per thread).
// Stores: amp_up (coalesced rows) + harmonic. HBM-store-bound (~133 MB).
// ---------------------------------------------------------------------------
__global__ void synth_kernel(const float* __restrict__ f0,        // [B,T]
                             const float* __restrict__ amp,       // [B,H,T]
                             const float* __restrict__ chunk_off, // [B,250]
                             float* __restrict__ out_harm,        // [B,TOUT]
                             float* __restrict__ out_ampup)       // [B,H,TOUT]
{
    const int chunk = blockIdx.x;   // 0..249
    const int b     = blockIdx.y;   // 0..7
    const int tid   = threadIdx.x;  // 0..255
    const int j     = chunk * HOP + tid;

    __shared__ float scan[256];
    __shared__ float samp[H_ * 4];  // 64 harmonics x 4 staged input samples

    // f0 linear upsample at j
    const float* f = f0 + b * T_;
    float pos = up_pos(j);
    int i0 = (int)pos;  if (i0 > T_ - 1) i0 = T_ - 1;
    int i1 = (i0 + 1 < T_) ? i0 + 1 : T_ - 1;
    float w  = pos - (float)i0;
    float fu = f[i0] * (1.0f - w) + f[i1] * w;

    // inclusive block scan of f0_up/SR  (phase in cycles)
    float v = fu * (1.0f / SRATE);
    scan[tid] = v;
    __syncthreads();
    for (int off = 1; off < 256; off <<= 1) {
        float t = (tid >= off) ? scan[tid - off] : 0.0f;
        __syncthreads();
        scan[tid] += t;
        __syncthreads();
    }
    float cyc = chunk_off[b * T_ + chunk] + scan[tid];
    float c   = cyc - floorf(cyc);   // sin(2*pi*n*cyc) == sin(2*pi*n*frac(cyc)), n integer

    // stage amp rows covering this chunk's interpolation span (<= 3 indices)
    int i_base = (int)up_pos(chunk * HOP);
    if (i_base > T_ - 1) i_base = T_ - 1;
    {
        const int h   = tid >> 2;
        const int s   = tid & 3;
        const int idx = i_base + s;
        samp[tid] = (idx < T_) ? amp[((size_t)b * H_ + h) * T_ + idx] : 0.0f;
    }
    __syncthreads();

    int o0 = i0 - i_base; o0 = o0 < 0 ? 0 : (o0 > 3 ? 3 : o0);
    int o1 = i1 - i_base; o1 = o1 < 0 ? 0 : (o1 > 3 ? 3 : o1);

    const float twopi = 6.283185307179586f;
    float accum = 0.0f;
    float* obase = out_ampup + ((size_t)b * H_) * TOUT + j;
    #pragma unroll 4
    for (int h = 0; h < H_; ++h) {
        float a0 = samp[h * 4 + o0];
        float a1 = samp[h * 4 + o1];
        float au = fmaf(w, a1 - a0, a0);
        obase[(size_t)h * TOUT] = au;                       // amp_up store (coalesced)
        accum += au * __sinf(twopi * (float)(h + 1) * c);   // v_sin_f32 (TRANS, co-exec)
    }
    out_harm[(size_t)b * TOUT + j] = accum;
}

// ---------------------------------------------------------------------------
extern "C" void kernel_launch(void* const* d_in, const int* in_sizes, int n_in,
                              void* d_out, int out_size, void* d_ws, size_t ws_size,
                              hipStream_t stream) {
    const float* z    = (const float*)d_in[0];   // [8,256,250]
    const float* f0   = (const float*)d_in[1];   // [8,1,250]
    const float* W    = (const float*)d_in[2];   // [64,256]
    const float* bias = (const float*)d_in[3];   // [64]

    float* out       = (float*)d_out;
    float* out_harm  = out;                           // [8,1,64000]
    float* out_ampup = out + (size_t)B_ * TOUT;       // [8,64,64000]

    float* amp       = (float*)d_ws;                  // [8,64,250]  (0.5 MB)
    float* chunk_off = amp + (size_t)B_ * H_ * T_;    // [8,250]

    amp_gemm_kernel<<<dim3(16, 4, 8), 32, 0, stream>>>(z, f0, W, bias, amp);
    chunk_scan_kernel<<<dim3(B_), 256, 0, stream>>>(f0, chunk_off);
    synth_kernel<<<dim3(T_, B_), 256, 0, stream>>>(f0, amp, chunk_off,
                                                   out_harm, out_ampup);
}